// GraphConvolution_35081292874421
// MI455X (gfx1250) — compile-verified
//
#include <hip/hip_runtime.h>
#include <hip/hip_bf16.h>

typedef float v2f __attribute__((ext_vector_type(2)));
typedef float v8f __attribute__((ext_vector_type(8)));

#define BATCH 512
#define NN    256
#define DIN   50
#define DOUT  50
#define DPAD  64
#define KCHUNK 128
#define LDS_STRIDE 76   // dwords; 2*76%64=24, 8*76%64=32 -> conflict-free B-frag reads & C-layout stores

// B fragment (4x16 f32) for WMMA_F32_16X16X4_F32 from LDS buffer with LDS_STRIDE row pitch.
// Assumed dual of the A 16x4 layout: lane = n + 16*(k/2), vgpr = k%2.
__device__ __forceinline__ v2f ld_bfrag(const float* __restrict__ s, int kbase, int j, int lane) {
    int r0 = kbase + 2 * (lane >> 4);
    int c  = 16 * j + (lane & 15);
    v2f bf;
    bf.x = s[(r0    ) * LDS_STRIDE + c];
    bf.y = s[(r0 + 1) * LDS_STRIDE + c];
    return bf;
}

__global__ __launch_bounds__(256)
void GraphConvolution_35081292874421_kernel(const float* __restrict__ nfm,
                                            const float* __restrict__ efm,
                                            const float* __restrict__ W,
                                            const float* __restrict__ bias,
                                            float* __restrict__ out) {
    // LDS: Ws = W^T zero-padded to 64x64 (pitch 76), Ys = 128x64 Y-chunk (pitch 76)
    __shared__ float smem[(DPAD + KCHUNK) * LDS_STRIDE];
    float* Ws = smem;
    float* Ys = smem + DPAD * LDS_STRIDE;

    const int tid  = threadIdx.x;
    const int lane = tid & 31;
    const int wave = tid >> 5;
    const int b    = blockIdx.x;
    const int m0   = blockIdx.y * 128 + wave * 16;   // this wave's 16 output rows

    // ---- Stage Ws[d][o] = W[o][d], zero-padded to 64x64 ----
    for (int i = tid; i < DPAD * DPAD; i += 256) {
        int d = i >> 6;
        int o = i & 63;
        float v = 0.0f;
        if (d < DIN && o < DOUT) v = W[o * DIN + d];
        Ws[d * LDS_STRIDE + o] = v;
    }
    __syncthreads();

    const int lcol  = lane & 15;
    const int lhalf = lane >> 4;

    v8f acc[4];
    #pragma unroll
    for (int j = 0; j < 4; ++j) acc[j] = (v8f)(0.0f);

    for (int kk = 0; kk < NN; kk += KCHUNK) {
        __syncthreads();  // previous chunk's Ys readers done before overwrite

        // ---- Y-phase: Y[kk+wave*16 .. +16][0..64) = Nf_rows @ W^T via WMMA ----
        v8f yacc[4];
        #pragma unroll
        for (int j = 0; j < 4; ++j) yacc[j] = (v8f)(0.0f);

        const int   mrow  = kk + wave * 16 + lcol;                     // Nf row (A-frag M = lane%16)
        const float* nfrow = nfm + ((size_t)b * NN + mrow) * DIN;

        #pragma unroll
        for (int kd = 0; kd < DPAD; kd += 4) {
            int d = kd + 2 * lhalf;                                    // even, so d<50 => d+1<=49 valid
            v2f a = {0.0f, 0.0f};
            if (d < DIN) a = *(const v2f*)(nfrow + d);
            #pragma unroll
            for (int j = 0; j < 4; ++j) {
                v2f bf = ld_bfrag(Ws, kd, j, lane);
                yacc[j] = __builtin_amdgcn_wmma_f32_16x16x4_f32(
                    false, a, false, bf, (short)0, yacc[j], false, false);
            }
        }

        // store Y slab (C-layout -> LDS rows)
        {
            int rbase = wave * 16 + 8 * lhalf;
            #pragma unroll
            for (int j = 0; j < 4; ++j) {
                #pragma unroll
                for (int r = 0; r < 8; ++r) {
                    Ys[(rbase + r) * LDS_STRIDE + 16 * j + lcol] = yacc[j][r];
                }
            }
        }
        __syncthreads();

        // ---- E-phase: acc += E[m0..m0+16][kk..kk+128] @ Ychunk ----
        const float* erow = efm + ((size_t)b * NN + m0 + lcol) * NN + kk + 2 * lhalf;
        #pragma unroll 8
        for (int k = 0; k < KCHUNK; k += 4) {
            v2f a = *(const v2f*)(erow + k);                           // global_load_b64, fully in-range
            #pragma unroll
            for (int j = 0; j < 4; ++j) {
                v2f bf = ld_bfrag(Ys, k, j, lane);
                acc[j] = __builtin_amdgcn_wmma_f32_16x16x4_f32(
                    false, a, false, bf, (short)0, acc[j], false, false);
            }
        }
    }

    // ---- Epilogue: bias + relu + masked store (C/D layout: row = m0 + r + 8*half, col = 16j+lcol) ----
    #pragma unroll
    for (int j = 0; j < 4; ++j) {
        int o = 16 * j + lcol;
        if (o < DOUT) {
            float bo = bias[o];
            #pragma unroll
            for (int r = 0; r < 8; ++r) {
                int n = m0 + r + 8 * lhalf;
                float v = acc[j][r] + bo;
                out[((size_t)b * NN + n) * DOUT + o] = v > 0.0f ? v : 0.0f;
            }
        }
    }
}

extern "C" void kernel_launch(void* const* d_in, const int* in_sizes, int n_in,
                              void* d_out, int out_size, void* d_ws, size_t ws_size,
                              hipStream_t stream) {
    const float* nfm  = (const float*)d_in[0];   // [512,256,50]
    const float* efm  = (const float*)d_in[1];   // [512,256,256]
    const float* W    = (const float*)d_in[2];   // [50,50]
    const float* bvec = (const float*)d_in[3];   // [50]
    float* out = (float*)d_out;                  // [512,256,50]

    dim3 grid(BATCH, 2, 1);
    dim3 block(256, 1, 1);
    GraphConvolution_35081292874421_kernel<<<grid, block, 0, stream>>>(nfm, efm, W, bvec, out);
}